// References_Rectifier_38362647888242
// MI455X (gfx1250) — compile-verified
//
#include <hip/hip_runtime.h>
#include <hip/hip_bf16.h>
#include <cstdint>

// Problem constants (match reference)
constexpr int Bn = 16;     // batch
constexpr int Nn = 256;    // tokens
constexpr int Cn = 768;    // dim
constexpr int Hn = 12;     // heads
constexpr int Dn = 64;     // head dim
constexpr int Ln = 10;     // classes
constexpr float SCALEC = 0.125f;  // 64^-0.5

typedef float v2f __attribute__((ext_vector_type(2)));
typedef float v8f __attribute__((ext_vector_type(8)));

static __device__ __forceinline__ v8f wmma_f32_16x16x4(v2f a, v2f b, v8f c) {
  // 8-arg pattern: (neg_a, A, neg_b, B, c_mod, C, reuse_a, reuse_b)
  return __builtin_amdgcn_wmma_f32_16x16x4_f32(false, a, false, b, (short)0, c,
                                               false, false);
}

// -----------------------------------------------------------------------------
// Row-major fp32 GEMM: C[M,N] = A[M,K] * B[K,N] (+ bias[N])
// Block = 256 threads (8 waves); block tile = 128(M) x 64(N).
// Each wave owns 16 rows and all 64 cols: 4 WMMA accumulators sharing one A
// fragment per K-step (A reuse x4, 4 independent WMMAs in flight).
// B panel (32 x 64) is double-buffered in LDS via CDNA5 async global->LDS
// copies (GLOBAL_LOAD_ASYNC_TO_LDS_B128 + s_wait_asynccnt).
// Grid: (N/64, M/128).  Requires M%128==0, N%64==0, K%32==0 (holds everywhere).
// -----------------------------------------------------------------------------
constexpr int KT = 32;   // K chunk staged per buffer

__global__ void gemm128x64_wmma(const float* __restrict__ A,
                                const float* __restrict__ Bm,
                                const float* __restrict__ bias,
                                float* __restrict__ Cm,
                                int M, int N, int K) {
  __shared__ float Blds[2][KT * 64];   // 2 x 8 KB

  const int tid   = threadIdx.x;
  const int lane  = tid & 31;
  const int wave  = tid >> 5;
  const int m0    = blockIdx.y * 128 + wave * 16;
  const int nblk  = blockIdx.x * 64;
  const int mrow  = lane & 15;          // A row / B col within 16-tile
  const int khalf = (lane >> 4) << 1;   // lanes 16..31 hold K=2,3

  const float* arow = A + (size_t)(m0 + mrow) * K;

  // Async-copy one 32x64 B chunk (8 KB) into Blds[buf]: 2 x b128 per lane.
  auto issue_copy = [&](int buf, int k0) {
    const int cg = (tid & 15) * 4;      // 4-float column group
    const int r0 = tid >> 4;            // 16 rows per issue
#pragma unroll
    for (int i = 0; i < 2; ++i) {
      const int row = i * 16 + r0;
      const float* g = Bm + (size_t)(k0 + row) * N + nblk + cg;
      const unsigned l =
          (unsigned)(uintptr_t)(&Blds[buf][row * 64 + cg]);  // LDS byte offset
      asm volatile("global_load_async_to_lds_b128 %0, %1, off"
                   :: "v"(l), "v"(g)
                   : "memory");
    }
  };

  const int nc = K / KT;
  issue_copy(0, 0);

  v8f acc[4] = {};
  for (int kc = 0; kc < nc; ++kc) {
    const int buf = kc & 1;
    if (kc + 1 < nc) {
      issue_copy(buf ^ 1, (kc + 1) * KT);
      // 2 ops of the fresh chunk may stay in flight; previous chunk is done.
      asm volatile("s_wait_asynccnt 0x2" ::: "memory");
    } else {
      asm volatile("s_wait_asynccnt 0x0" ::: "memory");
    }
    __syncthreads();                    // panel visible to the whole block

    const float* abase = arow + kc * KT;
    __builtin_prefetch(abase + KT, 0, 1);
    const float* Bb = &Blds[buf][0];
#pragma unroll
    for (int kk = 0; kk < KT; kk += 4) {
      v2f a;
      a.x = abase[kk + khalf];
      a.y = abase[kk + khalf + 1];
      const float* brow0 = Bb + (kk + khalf) * 64 + mrow;
      const float* brow1 = brow0 + 64;
      v2f b0, b1, b2, b3;
      b0.x = brow0[0];  b0.y = brow1[0];
      b1.x = brow0[16]; b1.y = brow1[16];
      b2.x = brow0[32]; b2.y = brow1[32];
      b3.x = brow0[48]; b3.y = brow1[48];
      acc[0] = wmma_f32_16x16x4(a, b0, acc[0]);
      acc[1] = wmma_f32_16x16x4(a, b1, acc[1]);
      acc[2] = wmma_f32_16x16x4(a, b2, acc[2]);
      acc[3] = wmma_f32_16x16x4(a, b3, acc[3]);
    }
    __syncthreads();                    // done reading buf before it refills
  }

  const int rbase = m0 + ((lane >= 16) ? 8 : 0);
#pragma unroll
  for (int j = 0; j < 4; ++j) {
    const int col = nblk + j * 16 + mrow;
    const float bv = bias ? bias[col] : 0.0f;
#pragma unroll
    for (int r = 0; r < 8; ++r)
      Cm[(size_t)(rbase + r) * N + col] = acc[j][r] + bv;
  }
}

// -----------------------------------------------------------------------------
// Per-(b,l,h) attention:
//   S[n,m] = -SCALE * sum_d q[b,n,h,d] * k[l,m,h,d]   (256x256, K=64) -> LDS
//   P = softmax over n (column-wise)                  -> attn out + LDS
//   T[n,d] = sum_m P[n,m] * v[l,m,h,d]                (256x64, K=256)
//   comb[b,n,h,d] += weights[b,h,l] * T[n,d]          (atomic f32)
// Block = 256 threads (8 waves); dynamic LDS = 256*257 floats (~257 KB; CDNA5
// WGP has 320 KB).  blockIdx.x = ((b*L + l)*H + h).
// -----------------------------------------------------------------------------
__global__ void attn_cls_kernel(const float* __restrict__ q,
                                const float* __restrict__ kmat,
                                const float* __restrict__ vmat,
                                const float* __restrict__ wts,
                                float* __restrict__ attn,
                                float* __restrict__ comb) {
  extern __shared__ float P[];            // [256][257] padded rows
  constexpr int SP = 257;

  const int bid  = blockIdx.x;
  const int h    = bid % Hn;
  const int l    = (bid / Hn) % Ln;
  const int b    = bid / (Hn * Ln);
  const int tid  = threadIdx.x;
  const int lane = tid & 31;
  const int wave = tid >> 5;
  const int mrow = lane & 15;
  const int khalf = (lane >> 4) << 1;

  const float* qh = q    + (size_t)b * Nn * Cn + h * Dn;   // row stride Cn
  const float* kh = kmat + (size_t)l * Nn * Cn + h * Dn;
  const float* vh = vmat + (size_t)l * Nn * Cn + h * Dn;

  // ---- Phase 1: scores into LDS (16x16 tiles; 256 tiles over 8 waves) ----
  for (int t = wave; t < 256; t += 8) {
    const int nt = (t >> 4) << 4;         // query-tile base
    const int mt = (t & 15) << 4;         // key-tile base
    const float* arow = qh + (size_t)(nt + mrow) * Cn;
    const float* brow = kh + (size_t)(mt + mrow) * Cn;
    v8f acc = {};
#pragma unroll
    for (int kk = 0; kk < Dn; kk += 4) {
      v2f a, bb;
      a.x  = arow[kk + khalf];
      a.y  = arow[kk + khalf + 1];
      bb.x = brow[kk + khalf];
      bb.y = brow[kk + khalf + 1];
      acc = wmma_f32_16x16x4(a, bb, acc);
    }
    const int col   = mt + mrow;
    const int rbase = nt + ((lane >= 16) ? 8 : 0);
#pragma unroll
    for (int r = 0; r < 8; ++r)
      P[(size_t)(rbase + r) * SP + col] = -SCALEC * acc[r];
  }
  __syncthreads();

  // ---- Phase 2: softmax over n for column m = tid; emit attn ----
  {
    const int m = tid;                    // 256 threads == 256 columns
    float mx = -3.4e38f;
    for (int n = 0; n < Nn; ++n) mx = fmaxf(mx, P[(size_t)n * SP + m]);
    float sum = 0.0f;
    for (int n = 0; n < Nn; ++n) {
      float e = __expf(P[(size_t)n * SP + m] - mx);
      P[(size_t)n * SP + m] = e;
      sum += e;
    }
    const float inv = 1.0f / sum;
    float* aout = attn + (size_t)bid * Nn * Nn;
    for (int n = 0; n < Nn; ++n) {
      float p = P[(size_t)n * SP + m] * inv;
      P[(size_t)n * SP + m] = p;
      aout[(size_t)n * Nn + m] = p;       // coalesced across lanes
    }
  }
  __syncthreads();

  // ---- Phase 3: T = P * v_h from LDS; weighted atomic accumulate ----
  const float w = wts[((size_t)b * Hn + h) * Ln + l];
  for (int t = wave; t < 64; t += 8) {    // 16 n-tiles x 4 d-tiles
    const int nt = (t >> 2) << 4;
    const int dt = (t & 3) << 4;
    const float* prow = &P[(size_t)(nt + mrow) * SP];
    v8f acc = {};
    for (int kk = 0; kk < Nn; kk += 4) {
      v2f a, bb;
      a.x  = prow[kk + khalf];
      a.y  = prow[kk + khalf + 1];
      bb.x = vh[(size_t)(kk + khalf) * Cn + dt + mrow];
      bb.y = vh[(size_t)(kk + khalf + 1) * Cn + dt + mrow];
      acc = wmma_f32_16x16x4(a, bb, acc);
    }
    const int dcol  = dt + mrow;
    const int rbase = nt + ((lane >= 16) ? 8 : 0);
#pragma unroll
    for (int r = 0; r < 8; ++r)
      atomicAdd(&comb[((size_t)b * Nn + rbase + r) * Cn + h * Dn + dcol],
                w * acc[r]);
  }
}

// -----------------------------------------------------------------------------
extern "C" void kernel_launch(void* const* d_in, const int* in_sizes, int n_in,
                              void* d_out, int out_size, void* d_ws,
                              size_t ws_size, hipStream_t stream) {
  const float* x       = (const float*)d_in[0];   // (16,256,768)
  const float* anchors = (const float*)d_in[1];   // (10,256,768)
  const float* weights = (const float*)d_in[2];   // (16,12,10)
  const float* W_qk    = (const float*)d_in[3];   // (768,768)
  const float* W_v     = (const float*)d_in[4];   // (768,768)
  const float* W_proj  = (const float*)d_in[5];   // (768,768)
  const float* b_proj  = (const float*)d_in[6];   // (768,)

  float* outp = (float*)d_out;                    // (B,N,C) = 3,145,728 floats
  float* attn = outp + (size_t)Bn * Nn * Cn;      // (B,L,H,N,N)

  // Workspace layout (floats): q | k | v | comb  (~41 MB total)
  float* q    = (float*)d_ws;                     // 4096 x 768
  float* kbuf = q    + (size_t)Bn * Nn * Cn;      // 2560 x 768
  float* vbuf = kbuf + (size_t)Ln * Nn * Cn;      // 2560 x 768
  float* comb = vbuf + (size_t)Ln * Nn * Cn;      // 4096 x 768 (accumulator)

  hipMemsetAsync(comb, 0, (size_t)Bn * Nn * Cn * sizeof(float), stream);

  dim3 blk(256);
  // q = x @ W_qk ; k = anchors @ W_qk ; v = anchors @ W_v
  gemm128x64_wmma<<<dim3(Cn / 64, (Bn * Nn) / 128), blk, 0, stream>>>(
      x, W_qk, nullptr, q, Bn * Nn, Cn, Cn);
  gemm128x64_wmma<<<dim3(Cn / 64, (Ln * Nn) / 128), blk, 0, stream>>>(
      anchors, W_qk, nullptr, kbuf, Ln * Nn, Cn, Cn);
  gemm128x64_wmma<<<dim3(Cn / 64, (Ln * Nn) / 128), blk, 0, stream>>>(
      anchors, W_v, nullptr, vbuf, Ln * Nn, Cn, Cn);

  // Attention + class-weighted combine
  const size_t shmem = (size_t)Nn * 257 * sizeof(float);  // ~257 KB (<320 KB WGP LDS)
  (void)hipFuncSetAttribute((const void*)attn_cls_kernel,
                            hipFuncAttributeMaxDynamicSharedMemorySize,
                            (int)shmem);
  attn_cls_kernel<<<dim3(Bn * Ln * Hn), blk, shmem, stream>>>(
      q, kbuf, vbuf, weights, attn, comb);

  // out = comb @ W_proj + b_proj
  gemm128x64_wmma<<<dim3(Cn / 64, (Bn * Nn) / 128), blk, 0, stream>>>(
      comb, W_proj, b_proj, outp, Bn * Nn, Cn, Cn);
}